// CBOW_46712064311622
// MI455X (gfx1250) — compile-verified
//
#include <hip/hip_runtime.h>
#include <math.h>

#define VOCAB   32000
#define DIM     128
#define BATCH   1024
#define CONTEXT 10

typedef float        v2f   __attribute__((ext_vector_type(2)));
typedef float        v8f   __attribute__((ext_vector_type(8)));
typedef unsigned int u32x4 __attribute__((ext_vector_type(4)));

// ---------------------------------------------------------------------------
// Kernel 1: scan the dense one-hot rows (1.31 GB, the bandwidth floor) and
// extract the index of the single non-zero element per (b,c) row.
// Non-temporal b128 loads so this stream does not flush w_out/emb out of L2.
// Exactly one lane finds a non-zero per row -> single writer, deterministic.
// ---------------------------------------------------------------------------
__global__ __launch_bounds__(256) void onehot_to_idx(const u32x4* __restrict__ batch,
                                                     int* __restrict__ idx) {
    const int row = blockIdx.x;                       // 0 .. B*C-1
    const u32x4* p = batch + (size_t)row * (VOCAB / 4);
    for (int i = threadIdx.x; i < VOCAB / 4; i += 256) {
        u32x4 v = __builtin_nontemporal_load(&p[i]);
        if (v.x | v.y | v.z | v.w) {
            int j = v.x ? 0 : (v.y ? 1 : (v.z ? 2 : 3));
            idx[row] = i * 4 + j;
        }
    }
}

// ---------------------------------------------------------------------------
// Kernel 2: avg_embedding[b][d] = (1/C) * sum_c emb[idx[b][c]][d]
// emb (16.4 MB) is L2-resident; each block gathers 10 rows coalesced.
// ---------------------------------------------------------------------------
__global__ __launch_bounds__(DIM) void avg_embed(const int* __restrict__ idx,
                                                 const float* __restrict__ emb,
                                                 float* __restrict__ avg) {
    const int b = blockIdx.x;
    const int d = threadIdx.x;
    float s = 0.f;
#pragma unroll
    for (int c = 0; c < CONTEXT; ++c) {
        const int ix = idx[b * CONTEXT + c];
        s += emb[(size_t)ix * DIM + d];
    }
    avg[b * DIM + d] = s * (1.0f / CONTEXT);
}

// ---------------------------------------------------------------------------
// Kernel 3: logits = avg[16xK=128] @ w_out[128xV], fused online softmax.
// One block = 16 batch rows, 8 waves; each wave owns N-tiles t = wave + 8k.
// f32 WMMA: V_WMMA_F32_16X16X4_F32, K-chain of 32 per 16x16 tile.
// Pass 1: online (max,sum) per lane -> shfl reduce -> LDS merge across waves.
// Pass 2: recompute tile, emit exp(x - rowMax) * rowInvSum with NT stores.
// ---------------------------------------------------------------------------
__global__ __launch_bounds__(256) void gemm_softmax(const float* __restrict__ avg,
                                                    const float* __restrict__ w_out,
                                                    float* __restrict__ out) {
    constexpr int LDSR = 132;                 // padded row stride: conflict-free b64 reads
    __shared__ float sA[16 * LDSR];
    __shared__ float sMax[8 * 16];
    __shared__ float sSum[8 * 16];
    __shared__ float rowMax[16];
    __shared__ float rowInv[16];

    const int tid   = threadIdx.x;
    const int wave  = tid >> 5;
    const int lane  = tid & 31;
    const int half  = lane >> 4;              // 0: lanes 0-15, 1: lanes 16-31
    const int n16   = lane & 15;
    const int bBase = blockIdx.x * 16;

    // Stage the 16x128 A tile (avg rows) into LDS.
    for (int i = tid; i < 16 * DIM; i += 256) {
        const int r = i >> 7, c = i & 127;
        sA[r * LDSR + c] = avg[(size_t)(bBase + r) * DIM + c];
    }
    __syncthreads();

    // Hoist per-lane A fragments for all 32 K-steps.
    // A 16x4 f32 layout: lane holds M = lane%16; VGPR0 = K {0|2}, VGPR1 = K {1|3} by half.
    v2f aReg[32];
#pragma unroll
    for (int ks = 0; ks < 32; ++ks) {
        const int k = ks * 4 + half * 2;
        aReg[ks].x = sA[n16 * LDSR + k];
        aReg[ks].y = sA[n16 * LDSR + k + 1];
    }

    // -------- Pass 1: GEMM + online (max, sum) ----------------------------
    float mx[8], sm[8];
#pragma unroll
    for (int r = 0; r < 8; ++r) { mx[r] = -3.4e38f; sm[r] = 0.f; }

    for (int t = wave; t < VOCAB / 16; t += 8) {
        const int col = t * 16 + n16;
        const float* bp = w_out + (size_t)(half * 2) * VOCAB + col;
        v8f acc = {0.f, 0.f, 0.f, 0.f, 0.f, 0.f, 0.f, 0.f};
#pragma unroll
        for (int ks = 0; ks < 32; ++ks) {
            v2f b;                             // B 4x16: lane holds N = lane%16, K {0|2},{1|3}
            b.x = bp[(size_t)(ks * 4) * VOCAB];
            b.y = bp[(size_t)(ks * 4 + 1) * VOCAB];
            acc = __builtin_amdgcn_wmma_f32_16x16x4_f32(
                false, aReg[ks], false, b, (short)0, acc, false, false);
        }
#pragma unroll
        for (int r = 0; r < 8; ++r) {          // D VGPR r -> row (r + 8*half), col n16
            const float x = acc[r];
            const float M = fmaxf(mx[r], x);
            sm[r] = sm[r] * __expf(mx[r] - M) + __expf(x - M);
            mx[r] = M;
        }
    }

    // Reduce over the 16 lanes of each half (xor masks 1..8 stay within the half).
#pragma unroll
    for (int r = 0; r < 8; ++r) {
        for (int off = 1; off <= 8; off <<= 1) {
            const float om = __shfl_xor(mx[r], off, 32);
            const float os = __shfl_xor(sm[r], off, 32);
            const float M  = fmaxf(mx[r], om);
            sm[r] = sm[r] * __expf(mx[r] - M) + os * __expf(om - M);
            mx[r] = M;
        }
    }
    if (n16 == 0) {
#pragma unroll
        for (int r = 0; r < 8; ++r) {
            sMax[wave * 16 + half * 8 + r] = mx[r];
            sSum[wave * 16 + half * 8 + r] = sm[r];
        }
    }
    __syncthreads();
    if (tid < 16) {                            // merge the 8 wave-partials per row
        float M = -3.4e38f;
        for (int w = 0; w < 8; ++w) M = fmaxf(M, sMax[w * 16 + tid]);
        float S = 0.f;
        for (int w = 0; w < 8; ++w) S += sSum[w * 16 + tid] * __expf(sMax[w * 16 + tid] - M);
        rowMax[tid] = M;
        rowInv[tid] = 1.0f / S;
    }
    __syncthreads();

    float rM[8], rI[8];
#pragma unroll
    for (int r = 0; r < 8; ++r) {
        rM[r] = rowMax[half * 8 + r];
        rI[r] = rowInv[half * 8 + r];
    }

    // -------- Pass 2: recompute GEMM, emit normalized softmax -------------
    for (int t = wave; t < VOCAB / 16; t += 8) {
        const int col = t * 16 + n16;
        const float* bp = w_out + (size_t)(half * 2) * VOCAB + col;
        v8f acc = {0.f, 0.f, 0.f, 0.f, 0.f, 0.f, 0.f, 0.f};
#pragma unroll
        for (int ks = 0; ks < 32; ++ks) {
            v2f b;
            b.x = bp[(size_t)(ks * 4) * VOCAB];
            b.y = bp[(size_t)(ks * 4 + 1) * VOCAB];
            acc = __builtin_amdgcn_wmma_f32_16x16x4_f32(
                false, aReg[ks], false, b, (short)0, acc, false, false);
        }
#pragma unroll
        for (int r = 0; r < 8; ++r) {
            const float e = __expf(acc[r] - rM[r]) * rI[r];
            __builtin_nontemporal_store(
                e, &out[(size_t)(bBase + half * 8 + r) * VOCAB + col]);
        }
    }
}

// ---------------------------------------------------------------------------
extern "C" void kernel_launch(void* const* d_in, const int* in_sizes, int n_in,
                              void* d_out, int out_size, void* d_ws, size_t ws_size,
                              hipStream_t stream) {
    const float* batch = (const float*)d_in[0];   // [B, C, V] one-hot f32
    const float* emb   = (const float*)d_in[1];   // [V, D] f32
    const float* w_out = (const float*)d_in[2];   // [D, V] f32
    float* out = (float*)d_out;                   // [B, V] f32

    // Workspace layout: idx[B*C] ints, then avg[B*D] floats (256B aligned).
    int*   idx = (int*)d_ws;
    size_t avgOff = ((size_t)BATCH * CONTEXT * sizeof(int) + 255) & ~(size_t)255;
    float* avg = (float*)((char*)d_ws + avgOff);

    onehot_to_idx<<<BATCH * CONTEXT, 256, 0, stream>>>((const u32x4*)batch, idx);
    avg_embed<<<BATCH, DIM, 0, stream>>>(idx, emb, avg);
    gemm_softmax<<<BATCH / 16, 256, 0, stream>>>(avg, w_out, out);
}